// PointMassModel_76897094467960
// MI455X (gfx1250) — compile-verified
//
#include <hip/hip_runtime.h>

// ---------------- types ----------------
typedef __attribute__((ext_vector_type(2))) float v2f;
typedef __attribute__((ext_vector_type(4))) float v4f;
typedef __attribute__((ext_vector_type(8))) float v8f;

struct WArg {
    // wt[n][f] = W[f][n]  (N-major so B fragments are K-contiguous in LDS)
    float wt[16][16];
};

#define WAVES_PER_BLOCK 8
#define TILES_PER_WAVE  8
#define ROW_STRIDE      18   // floats per LDS row (bank-conflict free, 8B aligned)

// ---------------- kernel ----------------
__launch_bounds__(256)
__global__ void pointmass_wmma_kernel(const float* __restrict__ X0,
                                      const float* __restrict__ U,
                                      float* __restrict__ Xf,
                                      int tiles, WArg warg) {
    constexpr int RS = ROW_STRIDE;
    __shared__ float ldsB[16 * RS];                       // W^T, shared by all waves
    __shared__ float ldsA[WAVES_PER_BLOCK][16 * RS];      // per-wave Z tile

    const int tid  = threadIdx.x;
    const int lane = tid & 31;
    const int wave = tid >> 5;

    // Stage W^T into LDS: thread t -> (n = t/16, f = t%16)
    {
        int n = tid >> 4, f = tid & 15;
        ldsB[n * RS + f] = warg.wt[n][f];
    }

    const int hl  = lane >> 4;   // lane half (0/1)
    const int l16 = lane & 15;
    // Fragment base (dwords): f32 16x4 A / 4x16 B WMMA layout:
    //   VGPR0 = K + 2*half, VGPR1 = K + 2*half + 1
    const int fragOff = l16 * RS + 2 * hl;

    float* zA = ldsA[wave];

    // ---- loop-invariant LDS scatter offsets (computed once) ----
    int offX[4], offXt[2], offU[4];
    #pragma unroll
    for (int e = 0; e < 4; ++e) {            // X main: g = 4*lane + e
        int g = 4 * lane + e;
        int r = g / 9;
        offX[e] = r * RS + (g - 9 * r);
    }
    #pragma unroll
    for (int e = 0; e < 2; ++e) {            // X tail (lane<8): g = 128 + 2*lane + e
        int g = 128 + 2 * lane + e;
        int r = g / 9;
        offXt[e] = r * RS + (g - 9 * r);
    }
    #pragma unroll
    for (int e = 0; e < 4; ++e) {            // U (lane<12): g = 4*lane + e
        int g = 4 * lane + e;
        int r = g / 3;
        offU[e] = r * RS + 9 + (g - 3 * r);
    }

    // ---- constant Z columns written once: col 12 = 1.0 (bias), 13..15 = 0 ----
    if (lane < 16) {
        v2f one0 = {1.0f, 0.0f};
        v2f zz   = {0.0f, 0.0f};
        *(v2f*)&zA[lane * RS + 12] = one0;
        *(v2f*)&zA[lane * RS + 14] = zz;
    }

    __syncthreads();   // ldsB visible to all waves

    // B fragments are loop-invariant: load once (ds_load_b64 pairs)
    v2f b0 = *(const v2f*)&ldsB[fragOff + 0];
    v2f b1 = *(const v2f*)&ldsB[fragOff + 4];
    v2f b2 = *(const v2f*)&ldsB[fragOff + 8];
    v2f b3 = *(const v2f*)&ldsB[fragOff + 12];

    const int waveGlobal = blockIdx.x * WAVES_PER_BLOCK + wave;
    const int tile0 = waveGlobal * TILES_PER_WAVE;

    // ---- per-lane global base pointers (per-iter deltas fold into IOFFSET) ----
    const float* xp  = X0 + (size_t)tile0 * 144 + 4 * lane;        // b128 stream
    const float* xtp = X0 + (size_t)tile0 * 144 + 128 + 2 * lane;  // tail (lane<8)
    const float* up  = U  + (size_t)tile0 * 48  + 4 * lane;        // (lane<12)
    float*       dp  = Xf + (size_t)tile0 * 144 + l16 + 72 * hl;   // output base

    // ---- software pipeline: preload tile 0 ----
    v4f x4; v2f x2; v4f u4;
    bool active = (tile0 < tiles);
    if (active) {
        x4 = *(const v4f*)(xp);
        if (lane < 8)  x2 = *(const v2f*)(xtp);
        if (lane < 12) u4 = *(const v4f*)(up);
    }

    #pragma unroll
    for (int it = 0; it < TILES_PER_WAVE; ++it) {
        // --- stage current tile registers into the wave-private LDS Z tile ---
        if (active) {
            zA[offX[0]] = x4[0]; zA[offX[1]] = x4[1];
            zA[offX[2]] = x4[2]; zA[offX[3]] = x4[3];
            if (lane < 8)  { zA[offXt[0]] = x2[0]; zA[offXt[1]] = x2[1]; }
            if (lane < 12) { zA[offU[0]] = u4[0]; zA[offU[1]] = u4[1];
                             zA[offU[2]] = u4[2]; zA[offU[3]] = u4[3]; }
        }
        __syncthreads();   // store->load visibility (uniform across waves)

        const bool curActive = active;

        // --- issue next tile's global loads early (overlap with WMMA+stores) ---
        if (it + 1 < TILES_PER_WAVE) {
            active = (tile0 + it + 1 < tiles);
            if (active) {
                x4 = *(const v4f*)(xp  + (it + 1) * 144);
                if (lane < 8)  x2 = *(const v2f*)(xtp + (it + 1) * 144);
                if (lane < 12) u4 = *(const v4f*)(up  + (it + 1) * 48);
            }
        }

        if (curActive) {
            // A fragments: K = 0,4,8,12 chunks (ds_load_b64 pairs)
            v2f a0 = *(const v2f*)&zA[fragOff + 0];
            v2f a1 = *(const v2f*)&zA[fragOff + 4];
            v2f a2 = *(const v2f*)&zA[fragOff + 8];
            v2f a3 = *(const v2f*)&zA[fragOff + 12];

            v8f d = {};
            d = __builtin_amdgcn_wmma_f32_16x16x4_f32(false, a0, false, b0, (short)0, d, false, false);
            d = __builtin_amdgcn_wmma_f32_16x16x4_f32(false, a1, false, b1, (short)0, d, false, false);
            d = __builtin_amdgcn_wmma_f32_16x16x4_f32(false, a2, false, b2, (short)0, d, false, false);
            d = __builtin_amdgcn_wmma_f32_16x16x4_f32(false, a3, false, b3, (short)0, d, false, false);

            // D layout: VGPR j -> env row (j + 8*half), col = l16 ; cols 0..8 valid
            if (l16 < 9) {
                #pragma unroll
                for (int j = 0; j < 8; ++j)
                    dp[it * 144 + j * 9] = d[j];
            }
        }
        // No trailing barrier needed: the Z tile is wave-private; the DS pipeline
        // is in-order per wave, so next iteration's ds_stores cannot pass this
        // iteration's ds_loads.
    }
}

// ---------------- host: closed-form affine map ----------------
static void m3mul(const double* A, const double* B, double* C) {
    for (int i = 0; i < 3; ++i)
        for (int j = 0; j < 3; ++j) {
            double s = 0.0;
            for (int k = 0; k < 3; ++k) s += A[i * 3 + k] * B[k * 3 + j];
            C[i * 3 + j] = s;
        }
}

extern "C" void kernel_launch(void* const* d_in, const int* in_sizes, int n_in,
                              void* d_out, int out_size, void* d_ws, size_t ws_size,
                              hipStream_t stream) {
    const float* X0 = (const float*)d_in[0];
    const float* U  = (const float*)d_in[1];
    float* Xf       = (float*)d_out;

    // ---- build the affine map on host (double precision) ----
    const double DT = 0.02, LMBDA = 0.5;
    const double c  = LMBDA / DT;          // 25
    const double h  = DT / 8.0;            // substep
    const double gz = -9.81;               // G_VEC.z

    double hA[9] = {0, h, 0,   0, 0, h,   0, 0, -c * h};
    double P2[9], P3[9], P4[9];
    m3mul(hA, hA, P2); m3mul(P2, hA, P3); m3mul(P3, hA, P4);
    const double I3[9] = {1,0,0, 0,1,0, 0,0,1};

    double R[9], S[9];
    for (int k = 0; k < 9; ++k) {
        R[k] = I3[k] + hA[k] + P2[k] / 2.0 + P3[k] / 6.0 + P4[k] / 24.0;
        S[k] = h * (I3[k] + hA[k] / 2.0 + P2[k] / 6.0 + P3[k] / 24.0);
    }
    double R2[9], R4[9], R8[9];
    m3mul(R, R, R2); m3mul(R2, R2, R4); m3mul(R4, R4, R8);
    // Sigma = I + R + ... + R^7 = (I+R)(I+R^2)(I+R^4)
    double IR[9], IR2[9], IR4[9], t1[9], Sig[9], T[9];
    for (int k = 0; k < 9; ++k) { IR[k] = I3[k] + R[k]; IR2[k] = I3[k] + R2[k]; IR4[k] = I3[k] + R4[k]; }
    m3mul(IR, IR2, t1); m3mul(t1, IR4, Sig);
    m3mul(Sig, S, T);   // X_final = R8 * X0_comp + T * b,  b = (0, -G_i, c*u_i)

    // ---- assemble W[16][16]: out[n] = sum_f Z[f] * W[f][n] ----
    double W[16][16] = {};
    for (int i = 0; i < 3; ++i) {              // xyz component
        for (int cs = 0; cs < 3; ++cs) {       // output block p/v/a
            int n = cs * 3 + i;
            for (int rs = 0; rs < 3; ++rs)     // input block p/v/a
                W[rs * 3 + i][n] = R8[cs * 3 + rs];
            W[9 + i][n] = c * T[cs * 3 + 2];                  // u coupling
            W[12][n]    = (i == 2) ? (-gz) * T[cs * 3 + 1]    // gravity bias (-G_z = +9.81)
                                   : 0.0;
        }
    }

    WArg warg;
    for (int n = 0; n < 16; ++n)
        for (int f = 0; f < 16; ++f)
            warg.wt[n][f] = (float)W[f][n];

    // ---- launch ----
    const int nEnv  = in_sizes[0] / 9;
    const int tiles = nEnv / 16;                                // 131072 for 2^21 envs
    const int tilesPerBlock = WAVES_PER_BLOCK * TILES_PER_WAVE; // 64
    const int blocks = (tiles + tilesPerBlock - 1) / tilesPerBlock;

    pointmass_wmma_kernel<<<blocks, 32 * WAVES_PER_BLOCK, 0, stream>>>(X0, U, Xf, tiles, warg);
}